// PointRendTrainPts2d_57947698757797
// MI455X (gfx1250) — compile-verified
//
#include <hip/hip_runtime.h>
#include <hip/hip_bf16.h>
#include <stdint.h>

// Problem constants (from reference)
#define NGRP 512
#define NCH  80
#define MH   64
#define MW   64
#define NS   2352   // oversampled points
#define NU   588    // uncertain (top-k) points
#define NR   196    // fresh random points
#define NP   784    // NU + NR
#define BLK  256

// Bilinear tap with zero padding outside [0,64)x[0,64)
__device__ __forceinline__ float tap_lds(const float* sm, int xi, int yi) {
    if ((unsigned)xi < (unsigned)MW && (unsigned)yi < (unsigned)MH)
        return sm[yi * MW + xi];
    return 0.0f;
}

__global__ __launch_bounds__(BLK)
void pointrend_train_pts_kernel(const float* __restrict__ in_map,
                                const int*   __restrict__ labels,
                                const float* __restrict__ rand_coords,
                                const float* __restrict__ rand_pts,
                                float*       __restrict__ out) {
    __shared__ __align__(16) float smap[MH * MW];  // 16 KB gt-channel map
    __shared__ __align__(16) float sunc[NS];       // 9.2 KB uncertainties

    const int g = blockIdx.x;
    const int t = threadIdx.x;

    // ---------------- Stage A: async global -> LDS copy of gt channel ----------------
    const int lab = labels[g];
    const float* gsrc = in_map + ((size_t)g * NCH + (size_t)lab) * (MH * MW);

    // Low 32 bits of the flat shared-pointer are the wave-relative LDS byte address.
    uint32_t lds0 = (uint32_t)(size_t)(void*)smap;
    // 16384 bytes total; 256 lanes * 16B = 4096B per issue; 4 issues.
#pragma unroll
    for (int k = 0; k < 4; ++k) {
        uint32_t byte_off = (uint32_t)(t * 16 + k * 4096);
        uint32_t lds_addr = lds0 + byte_off;
        // GVS mode: mem = SADDR + VADDR32 ; VDST = LDS byte address (ASYNCcnt)
        asm volatile("global_load_async_to_lds_b128 %0, %1, %2"
                     :
                     : "v"(lds_addr), "v"(byte_off), "s"(gsrc)
                     : "memory");
    }
    asm volatile("s_wait_asynccnt 0" ::: "memory");
    __syncthreads();

    // ---------------- Stage B: bilinear sample + uncertainty ----------------
    const float2* coords2 = (const float2*)(rand_coords) + (size_t)g * NS;
#pragma unroll 1
    for (int i = t; i < NS; i += BLK) {
        float2 c = coords2[i];
        float x = c.x * (float)MW - 0.5f;
        float y = c.y * (float)MH - 0.5f;
        float x0f = floorf(x);
        float y0f = floorf(y);
        float wx = x - x0f;
        float wy = y - y0f;
        int x0 = (int)x0f;
        int y0 = (int)y0f;
        float v00 = tap_lds(smap, x0,     y0);
        float v01 = tap_lds(smap, x0 + 1, y0);
        float v10 = tap_lds(smap, x0,     y0 + 1);
        float v11 = tap_lds(smap, x0 + 1, y0 + 1);
        float v = v00 * (1.0f - wx) * (1.0f - wy)
                + v01 * wx          * (1.0f - wy)
                + v10 * (1.0f - wx) * wy
                + v11 * wx          * wy;
        sunc[i] = -fabsf(v);
    }
    __syncthreads();

    // ---------------- Stage C: exact stable top-k by ranking ----------------
    // rank(i) = #{ j : u[j] > u[i]  or (u[j] == u[i] and j < i) }
    // Ranks are distinct; exactly NU points have rank < NU, and that rank is the
    // jax.lax.top_k output position (descending value, stable by index).
    const float4* sunc4 = (const float4*)sunc;  // NS/4 = 588 vectors
    float* out_g = out + (size_t)g * NP * 2;
#pragma unroll 1
    for (int i = t; i < NS; i += BLK) {
        const float ui = sunc[i];
        int rank = 0;
#pragma unroll 4
        for (int j4 = 0; j4 < NS / 4; ++j4) {
            float4 u = sunc4[j4];  // LDS broadcast (all lanes same addr)
            int j = j4 * 4;
            rank += (u.x > ui) || (u.x == ui && (j + 0) < i);
            rank += (u.y > ui) || (u.y == ui && (j + 1) < i);
            rank += (u.z > ui) || (u.z == ui && (j + 2) < i);
            rank += (u.w > ui) || (u.w == ui && (j + 3) < i);
        }
        if (rank < NU) {
            float2 c = coords2[i];
            out_g[(size_t)rank * 2 + 0] = c.x;
            out_g[(size_t)rank * 2 + 1] = c.y;
        }
    }

    // ---------------- Stage D: append fresh random points ----------------
    const float* rp = rand_pts + (size_t)g * NR * 2;
#pragma unroll 1
    for (int r = t; r < NR * 2; r += BLK) {
        out_g[NU * 2 + r] = rp[r];
    }
}

extern "C" void kernel_launch(void* const* d_in, const int* in_sizes, int n_in,
                              void* d_out, int out_size, void* d_ws, size_t ws_size,
                              hipStream_t stream) {
    (void)in_sizes; (void)n_in; (void)out_size; (void)d_ws; (void)ws_size;
    const float* in_map      = (const float*)d_in[0];   // [512, 80, 64, 64] f32
    const int*   labels      = (const int*)d_in[1];     // [512] i32
    const float* rand_coords = (const float*)d_in[2];   // [512, 2352, 2] f32
    const float* rand_pts    = (const float*)d_in[3];   // [512, 196, 2] f32
    float*       out         = (float*)d_out;           // [512, 784, 2] f32

    pointrend_train_pts_kernel<<<NGRP, BLK, 0, stream>>>(
        in_map, labels, rand_coords, rand_pts, out);
}